// FieldAwareFactorizationMachineLayer_69715909148812
// MI455X (gfx1250) — compile-verified
//
#include <hip/hip_runtime.h>

// FFM pairwise interaction: out[b,p,:] = x[b,i*N+j,:] * x[b,j*N+i,:] for i<j.
// B=2048, N=40, P=780 pairs, E=64 (f32). Pure streaming gather-multiply:
// ~1.23 GB moved, ~102 MFLOP -> memory bound at 23.3 TB/s (~53 us floor).
// No contraction dimension, so WMMA does not apply; optimize for HBM streaming.

#define N_FIELDS 40
#define NPAIRS   780        // 40*39/2
#define EMBED    64
#define BATCH    2048

struct PairTab { unsigned v[NPAIRS]; };

// Compile-time table: low16 = i*N+j (A row), high16 = j*N+i (B row).
constexpr PairTab makeTab() {
    PairTab t{};
    int p = 0;
    for (int i = 0; i < N_FIELDS; ++i)
        for (int j = i + 1; j < N_FIELDS; ++j) {
            t.v[p++] = (unsigned)(i * N_FIELDS + j) |
                       ((unsigned)(j * N_FIELDS + i) << 16);
        }
    return t;
}

__constant__ PairTab c_pairs = makeTab();

typedef float v4f __attribute__((ext_vector_type(4)));

__global__ __launch_bounds__(256) void ffm_pair_mul_kernel(
    const float* __restrict__ x, float* __restrict__ out) {
    // 8 threads per output row; each thread handles two float4 chunks (c, c+8)
    // -> 4 independent b128 loads + 2 b128 stores in flight per thread.
    unsigned tid = blockIdx.x * 256u + threadIdx.x;
    unsigned row = tid >> 3;          // output row in [0, BATCH*NPAIRS)
    unsigned c   = tid & 7u;          // float4 chunk id within the row
    unsigned b   = row / NPAIRS;      // magic-multiply division by 780
    unsigned p   = row - b * NPAIRS;

    unsigned pk = c_pairs.v[p];       // single cached 4B load
    unsigned ij = pk & 0xFFFFu;       // i*N+j
    unsigned ji = pk >> 16;           // j*N+i

    const size_t batch_base = (size_t)b * (N_FIELDS * N_FIELDS) * EMBED;
    const v4f* A  = (const v4f*)(x + batch_base + (size_t)ij * EMBED);
    const v4f* Bv = (const v4f*)(x + batch_base + (size_t)ji * EMBED);
    v4f*       O  = (v4f*)(out + (size_t)row * EMBED);

    // Non-temporal: zero reuse across the ~1.23 GB stream, keep it out of L2.
    v4f a0 = __builtin_nontemporal_load(A + c);
    v4f a1 = __builtin_nontemporal_load(A + c + 8);
    v4f b0 = __builtin_nontemporal_load(Bv + c);
    v4f b1 = __builtin_nontemporal_load(Bv + c + 8);

    __builtin_nontemporal_store(a0 * b0, O + c);
    __builtin_nontemporal_store(a1 * b1, O + c + 8);
}

extern "C" void kernel_launch(void* const* d_in, const int* in_sizes, int n_in,
                              void* d_out, int out_size, void* d_ws, size_t ws_size,
                              hipStream_t stream) {
    const float* x = (const float*)d_in[0];
    float* out = (float*)d_out;

    // 2048 * 780 rows * 8 threads/row = 12,779,520 threads = 49,920 blocks.
    const unsigned total_threads = (unsigned)BATCH * NPAIRS * 8u;
    dim3 grid(total_threads / 256u);
    dim3 block(256u);
    ffm_pair_mul_kernel<<<grid, block, 0, stream>>>(x, out);
}